// Symmetrizer_2843268350083
// MI455X (gfx1250) — compile-verified
//
#include <hip/hip_runtime.h>
#include <cstdint>

// ---------------------------------------------------------------------------
// Compile-time generation of the nu=2 / nu=3 contraction tables (MAX_L = 3),
// mirroring the Python reference exactly. Any mismatch in counts is a hard
// compile error (constexpr OOB / static_assert).
// ---------------------------------------------------------------------------
namespace tab {

constexpr int fact(int n) { int r = 1; for (int i = 2; i <= n; ++i) r *= i; return r; }

constexpr int lidx(int x, int y, int z) {
  int i = 0;
  for (int l = 0; l <= 3; ++l)
    for (int lx = l; lx >= 0; --lx)
      for (int ly = l - lx; ly >= 0; --ly) {
        const int lz = l - lx - ly;
        if (lx == x && ly == y && lz == z) return i;
        ++i;
      }
  return -1;
}

struct Nu2 { int i[19]; float p[19]; int k[19]; int n; };
constexpr Nu2 buildNu2() {
  Nu2 t{}; t.n = 0; int key = 0;
  for (int l = 1; l <= 3; ++l) {
    for (int ax = 0; ax <= l; ++ax)
      for (int ay = 0; ay <= l - ax; ++ay) {
        const int az = l - ax - ay;
        t.i[t.n] = lidx(ax, ay, az);
        t.p[t.n] = (float)(fact(l) / (fact(ax) * fact(ay) * fact(az)));
        t.k[t.n] = key; ++t.n;
      }
    ++key;
  }
  return t;
}

struct Nu3 { int i0[189]; int i1[189]; int i2[189]; float p[189]; int k[189]; int n; };
constexpr Nu3 buildNu3() {
  Nu3 t{}; t.n = 0; int key = 0;
  for (int k12 = 1; k12 <= 3; ++k12)
    for (int k23 = 1; k23 <= 3; ++k23)
      for (int k13 = 1; k13 <= 3; ++k13) {
        int m = k12 + k13;
        if (k12 + k23 > m) m = k12 + k23;
        if (k23 + k13 > m) m = k23 + k13;
        if (m > 3) continue;
        for (int ax = 0; ax <= k12; ++ax)
          for (int ay = 0; ay <= k12 - ax; ++ay) {
            const int az = k12 - ax - ay;
            const int pa = fact(k12) / (fact(ax) * fact(ay) * fact(az));
            for (int bx = 0; bx <= k23; ++bx)
              for (int by = 0; by <= k23 - bx; ++by) {
                const int bz = k23 - bx - by;
                const int pb = fact(k23) / (fact(bx) * fact(by) * fact(bz));
                for (int cx = 0; cx <= k13; ++cx)
                  for (int cy = 0; cy <= k13 - cx; ++cy) {
                    const int cz = k13 - cx - cy;
                    const int pc = fact(k13) / (fact(cx) * fact(cy) * fact(cz));
                    t.i0[t.n] = lidx(ax + cx, ay + cy, az + cz);
                    t.i1[t.n] = lidx(ax + bx, ay + by, az + bz);
                    t.i2[t.n] = lidx(bx + cx, by + cy, bz + cz);
                    t.p[t.n]  = (float)(pa * pb * pc);
                    t.k[t.n]  = key; ++t.n;
                  }
              }
          }
        ++key;
      }
  return t;
}

constexpr Nu2 NU2 = buildNu2();
constexpr Nu3 NU3 = buildNu3();
static_assert(NU2.n == 19,  "nu2 table count mismatch vs reference");
static_assert(NU3.n == 189, "nu3 table count mismatch vs reference");

} // namespace tab

// ---------------------------------------------------------------------------
// One block = 32 (n,r) tiles of 20x16 f32.
//   load : global -> LDS via global_load_async_to_lds_b128 (ASYNCcnt),
//          input tiles padded to 336 f32 for conflict-free column reads
//   math : fully unrolled register compute, 2 channels per thread
//   store: results staged to LDS (tile stride 144 f32 -> disjoint bank
//          groups), then global_store_async_from_lds_b128, 512 B contiguous
//          per wave-instruction; S_ENDPGM's implicit wait-idle drains ASYNCcnt
// ---------------------------------------------------------------------------
constexpr int TPB    = 256;   // threads per block (8 wave32)
constexpr int TILES  = 32;    // (n,r) tiles per block
constexpr int TSI    = 336;   // padded input tile stride in floats  (320 + 16)
constexpr int TSO    = 144;   // padded output tile stride in floats (128 + 16)

__global__ __launch_bounds__(TPB)
void symmetrizer_kernel(const float* __restrict__ in, float* __restrict__ out,
                        int nTiles) {
  __shared__ __align__(16) float smem [TILES * TSI];   // 43008 B input stage
  __shared__ __align__(16) float somem[TILES * TSO];   // 18432 B output stage

  const int tid = threadIdx.x;
  const long long tb = (long long)blockIdx.x * TILES;
  int th = nTiles - (int)tb;
  if (th > TILES) th = TILES;

  // ---- async stage in: global -> LDS, 16 B per lane per instruction -------
  const unsigned long long gbase = (unsigned long long)(in + tb * 320);
  const unsigned sbase = (unsigned)(unsigned long long)(const void*)&smem[0];
  const int nchunk = th * 80;                          // 80 x float4 per tile
#pragma unroll
  for (int j = 0; j < 10; ++j) {
    const int q = tid + j * TPB;
    if (q < nchunk) {
      const int t = q / 80;
      const int w = q - t * 80;
      const unsigned lo = sbase + (unsigned)(t * (TSI * 4) + w * 16);
      const unsigned go = (unsigned)(q * 16);
      asm volatile("global_load_async_to_lds_b128 %0, %1, %2"
                   :: "v"(lo), "v"(go), "s"(gbase) : "memory");
    }
  }
#if __has_builtin(__builtin_amdgcn_s_wait_asynccnt)
  __builtin_amdgcn_s_wait_asynccnt(0);
#else
  asm volatile("s_wait_asynccnt 0" ::: "memory");
#endif
  __syncthreads();

  // ---- per-thread column: tile lt, channel pair cp ------------------------
  const int lt = tid >> 3;      // 0..31
  const int cp = tid & 7;       // 0..7 -> channels 2cp, 2cp+1
  const float* sb = &smem[lt * TSI + cp * 2];

  float a0[20], a1[20];
#pragma unroll
  for (int l = 0; l < 20; ++l) {
    const float2 v = *(const float2*)(sb + l * 16);   // ds_load_2addr_b64
    a0[l] = v.x; a1[l] = v.y;
  }

  float o0[8], o1[8];
  o0[0] = a0[0]; o1[0] = a1[0];
#pragma unroll
  for (int k = 1; k < 8; ++k) { o0[k] = 0.f; o1[k] = 0.f; }

  // nu=2: b2[k] += p * a[i]^2      (19 terms, indices fold to constants)
#pragma unroll
  for (int t = 0; t < 19; ++t) {
    const int   i = tab::NU2.i[t];
    const float p = tab::NU2.p[t];
    const int   k = 1 + tab::NU2.k[t];
    o0[k] = __builtin_fmaf(p * a0[i], a0[i], o0[k]);
    o1[k] = __builtin_fmaf(p * a1[i], a1[i], o1[k]);
  }

  // nu=3: b3[k] += p * a[i0]*a[i1]*a[i2]   (189 terms; pair products CSE)
#pragma unroll
  for (int t = 0; t < 189; ++t) {
    const int   i0 = tab::NU3.i0[t], i1 = tab::NU3.i1[t], i2 = tab::NU3.i2[t];
    const float p  = tab::NU3.p[t];
    const int   k  = 4 + tab::NU3.k[t];
    const float t0 = a0[i0] * a0[i1];
    const float t1 = a1[i0] * a1[i1];
    o0[k] = __builtin_fmaf(p * t0, a0[i2], o0[k]);
    o1[k] = __builtin_fmaf(p * t1, a1[i2], o1[k]);
  }

  // ---- stage results to LDS (conflict-free: 144 mod 64 = 16) --------------
  if (lt < th) {
    float* so = &somem[lt * TSO + cp * 2];
#pragma unroll
    for (int k = 0; k < 8; ++k) {
      float2 v; v.x = o0[k]; v.y = o1[k];
      *(float2*)(so + k * 16) = v;                    // ds_store_b64
    }
  }
  // make the LDS staging visible to the async unit before other waves store
  asm volatile("s_wait_dscnt 0" ::: "memory");
  __syncthreads();

  // ---- async stage out: LDS -> global, 512 B contiguous per wave-instr ----
  const unsigned long long obase = (unsigned long long)(out + tb * 128);
  const unsigned obs = (unsigned)(unsigned long long)(const void*)&somem[0];
  const int nq = th * 32;                              // 32 x float4 per tile
#pragma unroll
  for (int j = 0; j < 4; ++j) {
    const int q = tid + j * TPB;
    if (q < nq) {
      const int t = q >> 5;
      const int w = q & 31;
      const unsigned lo = obs + (unsigned)(t * (TSO * 4) + w * 16);
      const unsigned go = (unsigned)(q * 16);
      asm volatile("global_store_async_from_lds_b128 %0, %1, %2"
                   :: "v"(go), "v"(lo), "s"(obase) : "memory");
    }
  }
  // S_ENDPGM performs an implicit wait-idle, draining ASYNCcnt.
}

// ---------------------------------------------------------------------------
extern "C" void kernel_launch(void* const* d_in, const int* in_sizes, int n_in,
                              void* d_out, int out_size, void* d_ws, size_t ws_size,
                              hipStream_t stream) {
  const float* in  = (const float*)d_in[0];
  float*       out = (float*)d_out;
  const int nTiles = in_sizes[0] / 320;               // (n,r) tiles of 20x16
  const int blocks = (nTiles + TILES - 1) / TILES;    // 32000/32 = 1000
  symmetrizer_kernel<<<blocks, TPB, 0, stream>>>(in, out, nTiles);
}